// MalwareGNN_25864293056851
// MI455X (gfx1250) — compile-verified
//
#include <hip/hip_runtime.h>

#define N_NODES 50000
#define N_EDGES 800000
#define EMB 256
#define GRAPHS 128

typedef float v2f __attribute__((ext_vector_type(2)));
typedef float v8f __attribute__((ext_vector_type(8)));

__global__ __launch_bounds__(256) void k_zero(float* __restrict__ p, int n) {
    int t = blockIdx.x * 256 + threadIdx.x;
    if (t < n) p[t] = 0.0f;
}

__global__ __launch_bounds__(256) void k_degree(const long long* __restrict__ dst,
                                                float* __restrict__ deg, int E) {
    int e = blockIdx.x * 256 + threadIdx.x;
    if (e < E) atomicAdd(&deg[(int)dst[e]], 1.0f);
}

__global__ __launch_bounds__(256) void k_dinv(float* __restrict__ d, int n) {
    int i = blockIdx.x * 256 + threadIdx.x;
    if (i < n) d[i] = rsqrtf(d[i] + 1.0f);  // +1 self-loop; always > 0
}

// C[M,Nmat] = A[M,128] @ B[128,Nmat]; one 16x16 tile per wave32 using
// V_WMMA_F32_16X16X4_F32 (full fp32 precision). A tile staged in LDS.
// blockIdx.x = M-tile, (blockIdx.y*8 + wave) = N-tile. M multiple of 16, Nmat mult of 128.
__global__ __launch_bounds__(256) void k_gemm_wmma(const float* __restrict__ A,
                                                   const float* __restrict__ B,
                                                   float* __restrict__ C, int Nmat) {
    __shared__ float As[16 * 128];
    const int tid = threadIdx.x;
    const long long row0 = (long long)blockIdx.x * 16;
    const float* Ablk = A + row0 * 128;
    #pragma unroll
    for (int idx = tid; idx < 2048; idx += 256) As[idx] = Ablk[idx];
    __syncthreads();

    const int wave = tid >> 5, lane = tid & 31;
    const int ln15 = lane & 15;        // = M for A-frag, = N for B-frag
    const int khalf = lane >> 4;       // lanes 16-31 hold K+2,K+3
    const int col0 = (blockIdx.y * 8 + wave) * 16;

    v8f acc = {0.f, 0.f, 0.f, 0.f, 0.f, 0.f, 0.f, 0.f};
    #pragma unroll 4
    for (int k = 0; k < 128; k += 4) {
        const int kk = k + 2 * khalf;
        v2f a = *(const v2f*)&As[ln15 * 128 + kk];            // ds_load_b64
        v2f b;
        b.x = B[(long long)kk * Nmat + col0 + ln15];
        b.y = B[(long long)(kk + 1) * Nmat + col0 + ln15];
        acc = __builtin_amdgcn_wmma_f32_16x16x4_f32(
            /*neg_a=*/false, a, /*neg_b=*/false, b,
            /*c_mod=*/(short)0, acc, /*reuse_a=*/false, /*reuse_b=*/false);
    }
    #pragma unroll
    for (int v = 0; v < 8; ++v) {
        const int m = v + 8 * khalf;                           // C/D layout §7.12.2
        C[(row0 + m) * (long long)Nmat + col0 + ln15] = acc[v];
    }
}

// out[i,:] = dinv[i]^2 * f[i,:]   (self-loop term; fully overwrites accumulator)
__global__ __launch_bounds__(256) void k_selfloop(const float* __restrict__ f,
                                                  const float* __restrict__ dinv,
                                                  float* __restrict__ out,
                                                  int fbits, int total) {
    int t = blockIdx.x * 256 + threadIdx.x;
    if (t >= total) return;
    float d = dinv[t >> fbits];
    out[t] = d * d * f[t];
}

// out[dst] += dinv[src]*dinv[dst] * feat[src]; 1 thread = 1 edge x 4 feats.
// Working set L2-resident (192 MB), so random gather/scatter stays on-chip.
__global__ __launch_bounds__(256) void k_edge_agg(const float4* __restrict__ feat,
                                                  const long long* __restrict__ src,
                                                  const long long* __restrict__ dst,
                                                  const float* __restrict__ dinv,
                                                  float* __restrict__ out,
                                                  int f4bits, int total) {
    int t = blockIdx.x * 256 + threadIdx.x;
    if (t >= total) return;
    int e = t >> f4bits;
    int j = t & ((1 << f4bits) - 1);
    int s = (int)src[e], d = (int)dst[e];
    float nrm = dinv[s] * dinv[d];
    float4 v = feat[(((long long)s) << f4bits) + j];
    float* o = out + ((((long long)d) << f4bits) + j) * 4LL;
    atomicAdd(o + 0, v.x * nrm);
    atomicAdd(o + 1, v.y * nrm);
    atomicAdd(o + 2, v.z * nrm);
    atomicAdd(o + 3, v.w * nrm);
}

__global__ __launch_bounds__(256) void k_bias_relu(float* __restrict__ h,
                                                   const float* __restrict__ b, int total) {
    int t = blockIdx.x * 256 + threadIdx.x;
    if (t >= total) return;
    float v = h[t] + b[t & 127];
    h[t] = fmaxf(v, 0.0f);
}

// sums[g,:] += h2[i,:] + b2  (b2 folded in: mean(h2+b2) == mean(h2)+b2)
__global__ __launch_bounds__(256) void k_pool(const float* __restrict__ h2,
                                              const long long* __restrict__ batch,
                                              const float* __restrict__ b2,
                                              float* __restrict__ sums,
                                              float* __restrict__ cnt) {
    int i = blockIdx.x, j = threadIdx.x;
    int g = (int)batch[i];
    atomicAdd(&sums[g * EMB + j], h2[(long long)i * EMB + j] + b2[j]);
    if (j == 0) atomicAdd(&cnt[g], 1.0f);
}

// mean -> L2 normalize -> @Wh + bh, one block per graph
__global__ __launch_bounds__(256) void k_final(const float* __restrict__ sums,
                                               const float* __restrict__ cnt,
                                               const float* __restrict__ Wh,
                                               const float* __restrict__ bh,
                                               float* __restrict__ out) {
    __shared__ float red[256];
    __shared__ float scale;
    int g = blockIdx.x, j = threadIdx.x;
    float inv = 1.0f / fmaxf(cnt[g], 1.0f);
    float e = sums[g * EMB + j] * inv;

    red[j] = e * e;
    __syncthreads();
    for (int s = 128; s > 0; s >>= 1) { if (j < s) red[j] += red[j + s]; __syncthreads(); }
    if (j == 0) scale = 1.0f / fmaxf(sqrtf(red[0]), 1e-12f);
    __syncthreads();
    float es = e * scale;

    red[j] = es * Wh[j * 2 + 0];
    __syncthreads();
    for (int s = 128; s > 0; s >>= 1) { if (j < s) red[j] += red[j + s]; __syncthreads(); }
    if (j == 0) out[g * 2 + 0] = red[0] + bh[0];
    __syncthreads();

    red[j] = es * Wh[j * 2 + 1];
    __syncthreads();
    for (int s = 128; s > 0; s >>= 1) { if (j < s) red[j] += red[j + s]; __syncthreads(); }
    if (j == 0) out[g * 2 + 1] = red[0] + bh[1];
}

extern "C" void kernel_launch(void* const* d_in, const int* in_sizes, int n_in,
                              void* d_out, int out_size, void* d_ws, size_t ws_size,
                              hipStream_t stream) {
    const float*     x     = (const float*)d_in[0];
    const long long* ei    = (const long long*)d_in[1];
    const long long* batch = (const long long*)d_in[2];
    const float*     W1    = (const float*)d_in[3];
    const float*     b1    = (const float*)d_in[4];
    const float*     W2    = (const float*)d_in[5];
    const float*     b2    = (const float*)d_in[6];
    const float*     Wh    = (const float*)d_in[7];
    const float*     bh    = (const float*)d_in[8];
    float*           out   = (float*)d_out;

    const long long* srcv = ei;            // edge_index[0]
    const long long* dstv = ei + N_EDGES;  // edge_index[1]

    // Workspace layout (floats), ~103 MB total with buffer reuse:
    float* ws   = (float*)d_ws;
    float* dinv = ws;                                   // [N]
    float* sums = ws + N_NODES;                         // [G*EMB]
    float* cnt  = sums + GRAPHS * EMB;                  // [G]
    float* R    = cnt + GRAPHS;                         // [N*256] (buf1|buf2, later agg2)
    float* buf1 = R;                                    // [N*128] xW1
    float* buf2 = R + (long long)N_NODES * 128;         // [N*128] agg1 -> h (relu)
    float* buf3 = R + (long long)N_NODES * 256;         // [N*256] h@W2

    dim3 b256(256);

    // degrees + pool accumulators (re-zeroed every call: no cross-call state)
    int zcount = N_NODES + GRAPHS * EMB + GRAPHS;
    k_zero  <<<dim3((zcount + 255) / 256), b256, 0, stream>>>(ws, zcount);
    k_degree<<<dim3((N_EDGES + 255) / 256), b256, 0, stream>>>(dstv, dinv, N_EDGES);
    k_dinv  <<<dim3((N_NODES + 255) / 256), b256, 0, stream>>>(dinv, N_NODES);

    // layer 1: xW1 -> D^-1/2 (A+I) D^-1/2 -> +b1, relu
    k_gemm_wmma<<<dim3(N_NODES / 16, 1), b256, 0, stream>>>(x, W1, buf1, 128);
    k_selfloop <<<dim3(N_NODES * 128 / 256), b256, 0, stream>>>(buf1, dinv, buf2, 7, N_NODES * 128);
    k_edge_agg <<<dim3(N_EDGES * 32 / 256), b256, 0, stream>>>((const float4*)buf1, srcv, dstv,
                                                               dinv, buf2, 5, N_EDGES * 32);
    k_bias_relu<<<dim3(N_NODES * 128 / 256), b256, 0, stream>>>(buf2, b1, N_NODES * 128);

    // layer 2: hW2 -> aggregation (b2 folded into pooling)
    k_gemm_wmma<<<dim3(N_NODES / 16, 2), b256, 0, stream>>>(buf2, W2, buf3, 256);
    k_selfloop <<<dim3(N_NODES * 256 / 256), b256, 0, stream>>>(buf3, dinv, R, 8, N_NODES * 256);
    k_edge_agg <<<dim3(N_EDGES * 64 / 256), b256, 0, stream>>>((const float4*)buf3, srcv, dstv,
                                                               dinv, R, 6, N_EDGES * 64);

    // mean-pool per graph, L2 normalize, head
    k_pool <<<dim3(N_NODES), b256, 0, stream>>>(R, batch, b2, sums, cnt);
    k_final<<<dim3(GRAPHS), b256, 0, stream>>>(sums, cnt, Wh, bh, out);
}